// AdaptiveFeatureRegularizer_62964220559918
// MI455X (gfx1250) — compile-verified
//
#include <hip/hip_runtime.h>
#include <stdint.h>

// ---------------------------------------------------------------------------
// AdaptiveFeatureRegularizer for MI455X (gfx1250, wave32)
// Memory-bound op (~0.5 GB traffic, ~0.3 GFLOP -> 0.6 FLOP/B): no GEMM, so no
// WMMA; ceiling is HBM at 23.3 TB/s (~21us). CDNA5-specific paths used:
//   - GLOBAL_LOAD_ASYNC_TO_LDS_B128 (ASYNCcnt) double-buffered staging of the
//     226MB feature stream (cp.async-style latency hiding, zero VGPR cost)
//   - s_wait_asynccnt / s_wait_dscnt split-counter synchronization
//   - non-temporal (th:NT) B128 stores for the 233MB streaming output
//   - global_prefetch_b8 two channels ahead
// Exact per-class 5%/95% quantiles via 4-pass MSD radix select on float bits
// (conf > 0 => uint bit order == float order). Deterministic (int atomics only).
// ---------------------------------------------------------------------------

typedef float        v4f __attribute__((ext_vector_type(4)));
typedef int          v4i __attribute__((ext_vector_type(4)));
typedef unsigned int v4u __attribute__((ext_vector_type(4)));

#define BB   2
#define CC   5
#define FFC  32
#define DHW  (96 * 96 * 96)          /* 884736  */
#define NVOX (BB * DHW)              /* 1769472 */
#define NG   (NVOX / 4)              /* 442368  */
#define NT   (CC * 4)                /* 20 radix-select targets */

// workspace layout (int32 indices into d_ws)
#define WS_COUNTS 0                  /* [5]  per-class voxel counts            */
#define WS_PREFIX 8                  /* [20] radix prefixes (become value bits) */
#define WS_RANK   32                 /* [20] remaining rank (-1 = invalid)     */
#define WS_FRACLO 64                 /* [5]  float: 5% interp fraction         */
#define WS_FRACHI 72                 /* [5]  float: 95% interp fraction        */
#define WS_TABLE  96                 /* [25] float: per-class coef table       */
#define WS_HIST   256                /* [20*256] global histograms             */
#define WS_TOTAL_INTS (256 + NT * 256)

__global__ void k_init(int* __restrict__ wsI) {
    for (int i = threadIdx.x; i < WS_TOTAL_INTS; i += blockDim.x) wsI[i] = 0;
}

// conf = 1 / sum_c exp(l_c/T - max_c(l_c/T));  also per-class voxel counts.
__global__ void k_conf(const float* __restrict__ logits,
                       const int*   __restrict__ labels,
                       float*       __restrict__ conf_out,
                       int*         __restrict__ wsI) {
    __shared__ int scnt[CC];
    const int tid = threadIdx.x;
    if (tid < CC) scnt[tid] = 0;
    __syncthreads();

    const int g = blockIdx.x * blockDim.x + tid;
    if (g < NG) {
        const int v0 = g * 4;
        const int b  = v0 / DHW;
        const int s  = v0 - b * DHW;
        const float* lp = logits + (size_t)b * CC * DHW + s;

        v4f x[CC];
        #pragma unroll
        for (int c = 0; c < CC; ++c)
            x[c] = (*(const v4f*)(lp + (size_t)c * DHW)) / 0.1f;  // logits / TEMPERATURE

        v4f conf;
        #pragma unroll
        for (int ln = 0; ln < 4; ++ln) {
            float mx = x[0][ln];
            #pragma unroll
            for (int c = 1; c < CC; ++c) mx = fmaxf(mx, x[c][ln]);
            float sum = 0.0f;
            #pragma unroll
            for (int c = 0; c < CC; ++c) sum += expf(x[c][ln] - mx);
            conf[ln] = 1.0f / sum;   // == max softmax prob
        }
        *(v4f*)(conf_out + v0) = conf;

        const v4i lb = *(const v4i*)(labels + v0);
        #pragma unroll
        for (int cl = 0; cl < CC; ++cl) {
            int cnt = (lb[0] == cl) + (lb[1] == cl) + (lb[2] == cl) + (lb[3] == cl);
            if (cnt) atomicAdd(&scnt[cl], cnt);
        }
    }
    __syncthreads();
    if (tid < CC && scnt[tid]) atomicAdd(&wsI[WS_COUNTS + tid], scnt[tid]);
}

// ranks & interpolation fractions for the 20 order statistics
__global__ void k_setup(int* __restrict__ wsI, float* __restrict__ wsF) {
    const int c = threadIdx.x;
    if (c >= CC) return;
    const int n = wsI[WS_COUNTS + c];
    int* rk = wsI + WS_RANK + c * 4;
    if (n > 0) {
        const float posLo = 0.05f * (float)(n - 1);
        const float posHi = 0.95f * (float)(n - 1);
        const int   kLo = (int)floorf(posLo);
        const int   kHi = (int)floorf(posHi);
        rk[0] = kLo; rk[1] = (kLo + 1 < n) ? kLo + 1 : n - 1;
        rk[2] = kHi; rk[3] = (kHi + 1 < n) ? kHi + 1 : n - 1;
        wsF[WS_FRACLO + c] = posLo - (float)kLo;
        wsF[WS_FRACHI + c] = posHi - (float)kHi;
    } else {
        rk[0] = rk[1] = rk[2] = rk[3] = -1;
        wsF[WS_FRACLO + c] = 0.0f;
        wsF[WS_FRACHI + c] = 0.0f;
    }
}

// one MSD radix pass: per-target 256-bucket histogram of byte `level`
__global__ void k_hist(const float* __restrict__ conf,
                       const int*   __restrict__ labels,
                       int*         __restrict__ wsI,
                       const int level) {
    __shared__ int      sh[NT * 256];
    __shared__ int      sRank[NT];
    __shared__ unsigned sPref[NT];
    const int tid = threadIdx.x;
    for (int i = tid; i < NT * 256; i += blockDim.x) sh[i] = 0;
    if (tid < NT) { sRank[tid] = wsI[WS_RANK + tid]; sPref[tid] = (unsigned)wsI[WS_PREFIX + tid]; }
    __syncthreads();

    const int shiftB = level * 8;
    for (int g = blockIdx.x * blockDim.x + tid; g < NG; g += gridDim.x * blockDim.x) {
        const int v0 = g * 4;
        const v4u u  = *(const v4u*)(conf + v0);    // positive floats: bit order == value order
        const v4i lb = *(const v4i*)(labels + v0);
        #pragma unroll
        for (int ln = 0; ln < 4; ++ln) {
            const unsigned bits = u[ln];
            const int cls = lb[ln];
            const unsigned hi = (level == 3) ? 0u : (bits >> ((level + 1) * 8));
            const int bucket = (int)((bits >> shiftB) & 255u);
            #pragma unroll
            for (int j = 0; j < 4; ++j) {
                const int t = cls * 4 + j;
                if (sRank[t] >= 0 && (level == 3 || hi == sPref[t]))
                    atomicAdd(&sh[t * 256 + bucket], 1);
            }
        }
    }
    __syncthreads();
    for (int i = tid; i < NT * 256; i += blockDim.x) {
        const int v = sh[i];
        if (v) atomicAdd(&wsI[WS_HIST + i], v);
    }
}

// select bucket containing the rank; refine prefix; zero hist for next pass
__global__ void k_scan(int* __restrict__ wsI) {
    const int t = threadIdx.x;
    if (t >= NT) return;
    int k = wsI[WS_RANK + t];
    if (k < 0) return;                       // empty class: hist row stayed zero
    int* h = wsI + WS_HIST + t * 256;
    int bucket = 255; bool found = false;
    for (int i = 0; i < 256; ++i) {
        const int c = h[i]; h[i] = 0;
        if (!found) { if (k < c) { found = true; bucket = i; } else k -= c; }
    }
    wsI[WS_PREFIX + t] = (wsI[WS_PREFIX + t] << 8) | bucket;
    wsI[WS_RANK + t]   = k;
}

// quantile interpolation + EMA + per-class coefficient table
__global__ void k_final(const float* __restrict__ gr,
                        const float* __restrict__ emin,
                        const float* __restrict__ emax,
                        const unsigned char* __restrict__ einit,
                        int* __restrict__ wsI, float* __restrict__ wsF) {
    if (threadIdx.x != 0 || blockIdx.x != 0) return;
    for (int c = 0; c < CC; ++c) {
        const int n = wsI[WS_COUNTS + c];
        float nmin, nmax;
        if (n > 0) {
            const float fLo = wsF[WS_FRACLO + c], fHi = wsF[WS_FRACHI + c];
            const float a0 = __int_as_float(wsI[WS_PREFIX + c * 4 + 0]);
            const float a1 = __int_as_float(wsI[WS_PREFIX + c * 4 + 1]);
            const float b0 = __int_as_float(wsI[WS_PREFIX + c * 4 + 2]);
            const float b1 = __int_as_float(wsI[WS_PREFIX + c * 4 + 3]);
            const float bmin = a0 * (1.0f - fLo) + a1 * fLo;   // linear interp (nanquantile)
            const float bmax = b0 * (1.0f - fHi) + b1 * fHi;
            const bool init = (einit[c] != 0);
            nmin = init ? 0.99f * emin[c] + 0.01f * bmin : bmin;
            nmax = init ? 0.99f * emax[c] + 0.01f * bmax : bmax;
        } else {                                   // has_vox == false
            nmin = emin[c]; nmax = emax[c];
        }
        const float inter = 1.0f - gr[c] / 4.0f;   // 1 - rank/(C-1)
        const bool  valid = (nmax > nmin);
        float* tb = wsF + WS_TABLE + c * 5;
        tb[0] = nmin;
        tb[1] = nmax;
        tb[2] = valid ? 1.0f / (nmax - nmin + 1e-8f) : 0.0f;
        tb[3] = inter;
        tb[4] = valid ? 1.0f : 0.0f;
    }
}

// ---------------------------------------------------------------------------
// Async global->LDS B128 copy (per-lane 16B), tracked by ASYNCcnt.
// lds_addr: wave-relative LDS byte address (low 32 bits of generic pointer).
// ---------------------------------------------------------------------------
__device__ __forceinline__ void async_ld16(unsigned lds_addr, const float* gaddr) {
    asm volatile("global_load_async_to_lds_b128 %0, %1, off"
                 :: "v"(lds_addr), "v"(gaddr)
                 : "memory");
}

// fused: gamma_map + feature scaling. Features stream through LDS via
// double-buffered async copies; one copy always in flight per wave.
__global__ void __launch_bounds__(256)
k_scale(const float* __restrict__ features,
        const int*   __restrict__ labels,
        const float* __restrict__ wsTab,
        float*       __restrict__ out_feat,
        float*       __restrict__ gamma_io) {    // holds conf on entry, gamma on exit
    __shared__ float tab[CC * 5];
    __shared__ v4f   stage[2][256];              // 2 x 4KB double buffer
    const int tid = threadIdx.x;
    if (tid < CC * 5) tab[tid] = wsTab[tid];
    __syncthreads();

    const int g  = blockIdx.x * blockDim.x + tid;   // grid is exact: g < NG always
    const int v0 = g * 4;
    const int b  = v0 / DHW;                        // DHW % 4 == 0: no batch straddle
    const int s  = v0 - b * DHW;
    const float* fbase = features + (size_t)b * FFC * DHW + s;
    float*       obase = out_feat + (size_t)b * FFC * DHW + s;

    const unsigned lds0 = (unsigned)(uintptr_t)(&stage[0][tid]);
    const unsigned lds1 = (unsigned)(uintptr_t)(&stage[1][tid]);

    async_ld16(lds0, fbase);                        // stage f=0; overlap with gamma math

    const v4f conf = *(const v4f*)(gamma_io + v0);
    const v4i lb   = *(const v4i*)(labels + v0);
    v4f gm;
    #pragma unroll
    for (int ln = 0; ln < 4; ++ln) {
        const int   base = lb[ln] * 5;
        const float cmin = tab[base + 0], cmax = tab[base + 1];
        const float inv  = tab[base + 2], inter = tab[base + 3], vld = tab[base + 4];
        const float cc = fminf(fmaxf(conf[ln], cmin), cmax);           // clip
        const float nc = (vld != 0.0f) ? (cc - cmin) * inv : 0.5f;     // normalized conf
        gm[ln] = 1.0f + inter * (1.0f + 0.5f * (1.0f - nc));           // gamma
    }
    __builtin_nontemporal_store(gm, (v4f*)(gamma_io + v0));

    #pragma unroll 2
    for (int f = 0; f < FFC; ++f) {
        const int cur = f & 1;
        if (f + 1 < FFC) {
            // WAR: async LDS writes are unordered vs DS reads -> retire the
            // previous consumer read of the buffer we are about to overwrite.
            asm volatile("s_wait_dscnt 0x0" ::: "memory");
            async_ld16(cur ? lds0 : lds1, fbase + (size_t)(f + 1) * DHW);
            asm volatile("s_wait_asynccnt 0x1" ::: "memory");  // older copy done
        } else {
            asm volatile("s_wait_asynccnt 0x0" ::: "memory");  // last copy done
        }
        if (f + 2 < FFC)
            __builtin_prefetch(fbase + (size_t)(f + 2) * DHW, 0, 1);   // global_prefetch_b8
        v4f d = stage[cur][tid];                    // ds_load_b128 (own lane's data)
        d *= gm;
        __builtin_nontemporal_store(d, (v4f*)(obase + (size_t)f * DHW));
    }
}

extern "C" void kernel_launch(void* const* d_in, const int* in_sizes, int n_in,
                              void* d_out, int out_size, void* d_ws, size_t ws_size,
                              hipStream_t stream) {
    const float*         features = (const float*)d_in[0];
    const float*         logits   = (const float*)d_in[1];
    const int*           labels   = (const int*)d_in[2];
    const float*         gr       = (const float*)d_in[3];
    const float*         emin     = (const float*)d_in[4];
    const float*         emax     = (const float*)d_in[5];
    const unsigned char* einit    = (const unsigned char*)d_in[6];

    float* out_feat = (float*)d_out;
    float* gamma_io = out_feat + (size_t)BB * FFC * DHW;   // conf scratch == gamma slot
    int*   wsI = (int*)d_ws;
    float* wsF = (float*)d_ws;

    k_init <<<1,      256, 0, stream>>>(wsI);
    k_conf <<<NG/256, 256, 0, stream>>>(logits, labels, gamma_io, wsI);
    k_setup<<<1,      32,  0, stream>>>(wsI, wsF);
    for (int lev = 3; lev >= 0; --lev) {
        k_hist<<<768, 256, 0, stream>>>(gamma_io, labels, wsI, lev);
        k_scan<<<1,   32,  0, stream>>>(wsI);
    }
    k_final<<<1,      32,  0, stream>>>(gr, emin, emax, einit, wsI, wsF);
    k_scale<<<NG/256, 256, 0, stream>>>(features, labels, wsF + WS_TABLE, out_feat, gamma_io);
}